// AttnDecoderRNN_11278584119922
// MI455X (gfx1250) — compile-verified
//
#include <hip/hip_runtime.h>
#include <math.h>

// ---------------------------------------------------------------------------
// AttnDecoderRNN single step, MI455X (gfx1250).
// Roofline: dominated by streaming input_enc + input_attW_enc (2 x 134 MB fp32)
// -> ~11.5us @ 23.3 TB/s. All GEMMs are M=64 (tiny, ~0.27 GMAC) -> done with
// V_WMMA_F32_16X16X4_F32 (full fp32, matches reference precision).
// h0==0 for all GRU cells => gh == b_hh, the *_hh matrices are never read.
// Softmax has no max-shift in the reference and normalization is linear =>
// single fused pass: accumulate unnormalized sum(w*enc) and sum(w).
// ---------------------------------------------------------------------------

typedef __attribute__((ext_vector_type(2))) float v2f;
typedef __attribute__((ext_vector_type(8))) float v8f;

#if __has_builtin(__builtin_amdgcn_tanhf)
__device__ __forceinline__ float fast_tanh(float x) { return __builtin_amdgcn_tanhf(x); }
#else
__device__ __forceinline__ float fast_tanh(float x) { return tanhf(x); }
#endif

__device__ __forceinline__ float fast_sigmoid(float x) { return 1.0f / (1.0f + expf(-x)); }

// ---------------------------------------------------------------------------
// Generic fp32 WMMA GEMM:  C[M,N] = act( A[M,K] @ W[N,Kw].T + bias )
// One 16x16 output tile per wave; K stepped by 4 via v_wmma_f32_16x16x4_f32.
// ksplit/kgap implement a K-column gather on W (for the x_att GEMM where the
// middle 512 columns of ga_ih multiply zeros: k>=ksplit reads col k+kgap).
// VGPR layouts per CDNA5 ISA 7.12.2:
//   A (16x4):  lane l(<16)/l+16 holds row M=l; vgpr j = K = k0 + 2*half + j
//   B (4x16):  vgpr j, half h   holds row K = k0 + j + 2*h, col N = n0 + l
//   C (16x16): vgpr i, half h   holds row M = m0 + i + 8*h, col N = n0 + l
// ---------------------------------------------------------------------------
__global__ __launch_bounds__(256) void gemm_wmma_f32(
    const float* __restrict__ A, int lda,
    const float* __restrict__ W, int ldw,
    const float* __restrict__ bias,
    float* __restrict__ C, int ldc,
    int M, int N, int K, int ksplit, int kgap, int act)
{
  const int lane = threadIdx.x & 31;
  const int wid  = threadIdx.x >> 5;
  const int l    = lane & 15;
  const int half = lane >> 4;

  const int ntn   = N >> 4;
  const int tiles = (M >> 4) * ntn;
  const int tile  = blockIdx.x * 8 + wid;
  if (tile >= tiles) return;              // wave-uniform: EXEC stays all-ones

  const int tn = tile % ntn;
  const int tm = tile / ntn;
  const int m0 = tm << 4;
  const int n0 = tn << 4;

  const float* __restrict__ arow = A + (size_t)(m0 + l) * lda;
  const float* __restrict__ wrow = W + (size_t)(n0 + l) * ldw;

  v8f c = {0.f, 0.f, 0.f, 0.f, 0.f, 0.f, 0.f, 0.f};
  for (int k0 = 0; k0 < K; k0 += 4) {
    const int k   = k0 + 2 * half;
    const int kw0 = (k     < ksplit) ? k     : k + kgap;
    const int kw1 = (k + 1 < ksplit) ? k + 1 : k + 1 + kgap;
    v2f a, b;
    a[0] = arow[k];
    a[1] = arow[k + 1];
    b[0] = wrow[kw0];
    b[1] = wrow[kw1];
    c = __builtin_amdgcn_wmma_f32_16x16x4_f32(false, a, false, b, (short)0, c,
                                              false, false);
  }

  const float bv = bias ? bias[n0 + l] : 0.f;
#pragma unroll
  for (int i = 0; i < 8; ++i) {
    const int row = m0 + i + 8 * half;
    float v = c[i] + bv;
    if (act == 1) v = fmaxf(v, 0.f);
    C[(size_t)row * ldc + (n0 + l)] = v;
  }
}

// ---------------------------------------------------------------------------
// GRU combine with h == 0:
//   r = sig(gi_r + bhh_r); z = sig(gi_z + bhh_z)
//   n = tanh(gi_n + r*bhh_n);  out = addin + (1-z)*n       (z*h term == 0)
// gi is [64,1536] (bih already added by the GEMM); out is [64,512].
// ---------------------------------------------------------------------------
__global__ __launch_bounds__(256) void gru_combine(
    const float* __restrict__ gi, const float* __restrict__ bhh,
    const float* __restrict__ addin, float* __restrict__ out)
{
  const int idx = blockIdx.x * blockDim.x + threadIdx.x;   // 64*512
  const int n = idx >> 9, j = idx & 511;
  const float* g = gi + (size_t)n * 1536;
  const float r  = fast_sigmoid(g[j]        + bhh[j]);
  const float z  = fast_sigmoid(g[512 + j]  + bhh[512 + j]);
  const float nn = fast_tanh   (g[1024 + j] + r * bhh[1024 + j]);
  const float v  = (1.0f - z) * nn;
  out[idx] = (addin ? addin[idx] : 0.0f) + v;
}

// Zero the attention accumulators; scatter style_vec into x_eff cols [256,384).
__global__ __launch_bounds__(256) void prep_kernel(
    float* __restrict__ acc, float* __restrict__ sumw,
    const float* __restrict__ sv, float* __restrict__ xeff)
{
  const int idx = blockIdx.x * blockDim.x + threadIdx.x;
  if (idx < 64 * 512) acc[idx] = 0.f;
  if (idx < 64)       sumw[idx] = 0.f;
  if (idx < 64 * 128) {
    const int n = idx >> 7, d = idx & 127;
    xeff[n * 384 + 256 + d] = sv[idx];
  }
}

// ---------------------------------------------------------------------------
// Fused attention stream: one pass over both 134 MB tensors.
// grid = (T/TCHUNK, N), block = 256 (8 waves); wave w owns t = t0+w, t0+w+8,...
// Per t: score = sum_d wattn[d]*tanh(awenc[n,t,d]+attWdec[n,d]) + battn
//        w = (t<len) ? exp(score) : 0
//        acc += w * enc[n,t,:]; sw += w        (normalize later)
// Lane l owns d = 4*l + 128*j (j=0..3): fully coalesced b128 loads.
// Wave-private register accumulators -> ds_add_f32 block reduce ->
// one global_atomic_add_f32 set per block.
// ---------------------------------------------------------------------------
#define TCHUNK 128

__global__ __launch_bounds__(256) void attn_stream(
    const float* __restrict__ enc,      // [N,T,512]
    const float* __restrict__ awenc,    // [N,T,512]
    const float* __restrict__ attwdec,  // [N,512]
    const float* __restrict__ wattn,    // [512]
    const float* __restrict__ battn,    // [1]
    const int*   __restrict__ lengths,  // [N]
    float* __restrict__ acc,            // [N,512] unnormalized
    float* __restrict__ sumw,           // [N]
    int T)
{
  __shared__ float s_aw[512];
  __shared__ float s_wt[512];
  __shared__ float s_acc[512];
  __shared__ float s_sw;

  const int n    = blockIdx.y;
  const int t0   = blockIdx.x * TCHUNK;
  const int tid  = threadIdx.x;
  const int lane = tid & 31;
  const int wid  = tid >> 5;

  s_aw[tid]       = attwdec[n * 512 + tid];
  s_aw[tid + 256] = attwdec[n * 512 + tid + 256];
  s_wt[tid]       = wattn[tid];
  s_wt[tid + 256] = wattn[tid + 256];
  s_acc[tid] = 0.f;
  s_acc[tid + 256] = 0.f;
  if (tid == 0) s_sw = 0.f;
  __syncthreads();

  const int   len = lengths[n];
  const float bt  = battn[0];
  const int   d0  = lane * 4;

  float4 a0 = {0,0,0,0}, a1 = {0,0,0,0}, a2 = {0,0,0,0}, a3 = {0,0,0,0};
  float  sw = 0.f;

  for (int t = t0 + wid; t < t0 + TCHUNK; t += 8) {
    const float* __restrict__ xp = awenc + ((size_t)n * T + t) * 512;
    const float* __restrict__ ep = enc   + ((size_t)n * T + t) * 512;
    __builtin_prefetch(xp + 8 * 512 + d0, 0, 0);   // global_prefetch_b8

    const float4 x0 = *(const float4*)(xp + d0);
    const float4 x1 = *(const float4*)(xp + d0 + 128);
    const float4 x2 = *(const float4*)(xp + d0 + 256);
    const float4 x3 = *(const float4*)(xp + d0 + 384);

    float p = 0.f;
#define TERM(xv, dd)                                          \
    p += s_wt[(dd) + 0] * fast_tanh((xv).x + s_aw[(dd) + 0]); \
    p += s_wt[(dd) + 1] * fast_tanh((xv).y + s_aw[(dd) + 1]); \
    p += s_wt[(dd) + 2] * fast_tanh((xv).z + s_aw[(dd) + 2]); \
    p += s_wt[(dd) + 3] * fast_tanh((xv).w + s_aw[(dd) + 3]);
    TERM(x0, d0) TERM(x1, d0 + 128) TERM(x2, d0 + 256) TERM(x3, d0 + 384)
#undef TERM

    // wave32 all-reduce
#pragma unroll
    for (int off = 16; off > 0; off >>= 1) p += __shfl_xor(p, off, 32);

    if (t < len) {                       // wave-uniform: skip masked rows
      const float wgt = expf(p + bt);
      __builtin_prefetch(ep + 8 * 512 + d0, 0, 0);
      const float4 e0 = *(const float4*)(ep + d0);
      const float4 e1 = *(const float4*)(ep + d0 + 128);
      const float4 e2 = *(const float4*)(ep + d0 + 256);
      const float4 e3 = *(const float4*)(ep + d0 + 384);
      a0.x += wgt * e0.x; a0.y += wgt * e0.y; a0.z += wgt * e0.z; a0.w += wgt * e0.w;
      a1.x += wgt * e1.x; a1.y += wgt * e1.y; a1.z += wgt * e1.z; a1.w += wgt * e1.w;
      a2.x += wgt * e2.x; a2.y += wgt * e2.y; a2.z += wgt * e2.z; a2.w += wgt * e2.w;
      a3.x += wgt * e3.x; a3.y += wgt * e3.y; a3.z += wgt * e3.z; a3.w += wgt * e3.w;
      if (lane == 0) sw += wgt;
    }
  }

  // per-wave partials -> LDS (ds_add_f32)
  atomicAdd(&s_acc[d0 + 0], a0.x); atomicAdd(&s_acc[d0 + 1], a0.y);
  atomicAdd(&s_acc[d0 + 2], a0.z); atomicAdd(&s_acc[d0 + 3], a0.w);
  atomicAdd(&s_acc[d0 + 128], a1.x); atomicAdd(&s_acc[d0 + 129], a1.y);
  atomicAdd(&s_acc[d0 + 130], a1.z); atomicAdd(&s_acc[d0 + 131], a1.w);
  atomicAdd(&s_acc[d0 + 256], a2.x); atomicAdd(&s_acc[d0 + 257], a2.y);
  atomicAdd(&s_acc[d0 + 258], a2.z); atomicAdd(&s_acc[d0 + 259], a2.w);
  atomicAdd(&s_acc[d0 + 384], a3.x); atomicAdd(&s_acc[d0 + 385], a3.y);
  atomicAdd(&s_acc[d0 + 386], a3.z); atomicAdd(&s_acc[d0 + 387], a3.w);
  if (lane == 0) atomicAdd(&s_sw, sw);
  __syncthreads();

  // block partial -> global (global_atomic_add_f32)
  atomicAdd(&acc[n * 512 + tid], s_acc[tid]);
  atomicAdd(&acc[n * 512 + tid + 256], s_acc[tid + 256]);
  if (tid == 0) atomicAdd(&sumw[n], s_sw);
}

// Normalize attention, emit attn_applied output, assemble out_dec = [attn,out_att,sv].
__global__ __launch_bounds__(256) void finalize_attn(
    const float* __restrict__ acc, const float* __restrict__ sumw,
    const float* __restrict__ outatt, const float* __restrict__ sv,
    float* __restrict__ attn_out,      // d_out + 64*160
    float* __restrict__ outdec)        // [64,1152]
{
  const int idx = blockIdx.x * blockDim.x + threadIdx.x;   // 64*512
  const int n = idx >> 9, d = idx & 511;
  const float s  = fmaxf(sumw[n], 1e-12f);
  const float av = acc[idx] / s;
  attn_out[idx] = av;
  float* od = outdec + (size_t)n * 1152;
  od[d]       = av;
  od[512 + d] = outatt[idx];
  if (d < 128) od[1024 + d] = sv[n * 128 + d];
}

// ---------------------------------------------------------------------------
static inline void launch_gemm(hipStream_t s, const float* A, int lda,
                               const float* W, int ldw, const float* bias,
                               float* C, int ldc, int N, int K,
                               int ksplit, int kgap, int act)
{
  const int tiles  = 4 * (N >> 4);               // M = 64 -> 4 row tiles
  const int blocks = (tiles + 7) / 8;
  hipLaunchKernelGGL(gemm_wmma_f32, dim3(blocks), dim3(256), 0, s,
                     A, lda, W, ldw, bias, C, ldc, 64, N, K, ksplit, kgap, act);
}

extern "C" void kernel_launch(void* const* d_in, const int* in_sizes, int n_in,
                              void* d_out, int out_size, void* d_ws, size_t ws_size,
                              hipStream_t stream)
{
  (void)in_sizes; (void)n_in; (void)out_size; (void)ws_size;
  const float* input_enc   = (const float*)d_in[0];
  const float* input_awenc = (const float*)d_in[1];
  const float* input_dec   = (const float*)d_in[2];
  const float* style_vec   = (const float*)d_in[3];
  const int*   lengths     = (const int*)  d_in[4];
  const float* pw1    = (const float*)d_in[5];
  const float* pb1    = (const float*)d_in[6];
  const float* pw2    = (const float*)d_in[7];
  const float* pb2    = (const float*)d_in[8];
  const float* wdec   = (const float*)d_in[9];
  const float* bdec   = (const float*)d_in[10];
  const float* ga_ih  = (const float*)d_in[11];
  const float* ga_bih = (const float*)d_in[13];
  const float* ga_bhh = (const float*)d_in[14];
  const float* wattn  = (const float*)d_in[15];
  const float* battn  = (const float*)d_in[16];
  const float* wsc    = (const float*)d_in[17];
  const float* bsc    = (const float*)d_in[18];
  const float* g1_ih  = (const float*)d_in[19];
  const float* g1_bih = (const float*)d_in[21];
  const float* g1_bhh = (const float*)d_in[22];
  const float* g2_ih  = (const float*)d_in[23];
  const float* g2_bih = (const float*)d_in[25];
  const float* g2_bhh = (const float*)d_in[26];
  const float* wout   = (const float*)d_in[27];
  const float* bout   = (const float*)d_in[28];
  // ga_hh / g1_hh / g2_hh intentionally unused: h0 == 0 => gh == b_hh.

  constexpr int T = 1024;

  float* ws        = (float*)d_ws;
  float* ws_pre1   = ws;                   // 64*512
  float* ws_xeff   = ws_pre1   + 64*512;   // 64*384  (pre cols [0,256) + sv [256,384))
  float* ws_gi     = ws_xeff   + 64*384;   // 64*1536 (reused for gi_att and gi1)
  float* ws_outatt = ws_gi     + 64*1536;  // 64*512
  float* ws_attw   = ws_outatt + 64*512;   // 64*512
  float* ws_acc    = ws_attw   + 64*512;   // 64*512
  float* ws_sumw   = ws_acc    + 64*512;   // 64
  float* ws_outdec = ws_sumw   + 64;       // 64*1152
  float* ws_res    = ws_outdec + 64*1152;  // 64*512
  float* ws_gi2    = ws_res    + 64*512;   // 64*1536

  float* out_main = (float*)d_out;          // [64,2,80] row-major = [64,160]
  float* out_attn = out_main + 64 * 160;    // [64,1,512]

  // 0) zero accumulators + scatter style into x_eff tail
  hipLaunchKernelGGL(prep_kernel, dim3(128), dim3(256), 0, stream,
                     ws_acc, ws_sumw, style_vec, ws_xeff);

  // Phase A: pre-net + attention GRU + attW_dec (all WMMA)
  launch_gemm(stream, input_dec, 80,  pw1, 80,  pb1, ws_pre1, 512, 512, 80,   80,   0, 1);
  launch_gemm(stream, ws_pre1, 512,  pw2, 512, pb2, ws_xeff, 384, 256, 512,  512,  0, 1);
  // gi_att: K=384 effective; W cols k>=256 map to ga_ih col k+512 (skip zero ctx)
  launch_gemm(stream, ws_xeff, 384,  ga_ih, 896, ga_bih, ws_gi, 1536, 1536, 384, 256, 512, 0);
  hipLaunchKernelGGL(gru_combine, dim3(128), dim3(256), 0, stream,
                     ws_gi, ga_bhh, (const float*)nullptr, ws_outatt);
  launch_gemm(stream, ws_outatt, 512, wdec, 512, bdec, ws_attw, 512, 512, 512, 512, 0, 0);

  // Phase B: fused tanh-score-softmax-weighted-sum over the two 134 MB tensors
  hipLaunchKernelGGL(attn_stream, dim3(T / TCHUNK, 64), dim3(256), 0, stream,
                     input_enc, input_awenc, ws_attw, wattn, battn, lengths,
                     ws_acc, ws_sumw, T);

  // Phase C: normalize + decoder stack
  hipLaunchKernelGGL(finalize_attn, dim3(128), dim3(256), 0, stream,
                     ws_acc, ws_sumw, ws_outatt, style_vec, out_attn, ws_outdec);
  launch_gemm(stream, ws_outdec, 1152, wsc,   1152, bsc,    ws_res, 512, 512, 1152, 1152, 0, 0);
  launch_gemm(stream, ws_outdec, 1152, g1_ih, 1152, g1_bih, ws_gi, 1536, 1536, 1152, 1152, 0, 0);
  hipLaunchKernelGGL(gru_combine, dim3(128), dim3(256), 0, stream,
                     ws_gi, g1_bhh, ws_res, ws_res);
  launch_gemm(stream, ws_res, 512, g2_ih, 512, g2_bih, ws_gi2, 1536, 1536, 512, 512, 0, 0);
  hipLaunchKernelGGL(gru_combine, dim3(128), dim3(256), 0, stream,
                     ws_gi2, g2_bhh, ws_res, ws_res);
  launch_gemm(stream, ws_res, 512, wout, 512, bout, out_main, 160, 160, 512, 512, 0, 0);
}